// LearnableUpsampling_12438225289585
// MI455X (gfx1250) — compile-verified
//
#include <hip/hip_runtime.h>
#include <math.h>

// ---------------- constants ----------------
#define B_   16
#define TS_  512
#define D_   192
#define CO_  8
#define DW_  4
#define DC_  2
#define TM_  1000

// output layout (floats), return order: out, ~mel_mask, mel_len, Wp, C
#define OUT_OFF   0
#define OUT_N     (B_*TM_*2*D_)          // 6,144,000
#define MASK_OFF  (OUT_OFF + OUT_N)      // 6,144,000
#define MASK_N    (B_*TM_)               // 16,000
#define MEL_OFF   (MASK_OFF + MASK_N)    // 6,160,000
#define MEL_N     (B_)                   // 16
#define WP_OFF    (MEL_OFF + MEL_N)      // 6,160,016
#define WP_N      ((size_t)B_*DW_*TM_*TS_)
#define C_OFF     (WP_OFF + (size_t)WP_N)
#define C_N       ((size_t)B_*TM_*TS_*DC_)

typedef __attribute__((ext_vector_type(2))) float v2f;
typedef __attribute__((ext_vector_type(8))) float v8f;

__device__ __forceinline__ float silu_f(float a) { return a / (1.f + expf(-a)); }

// ---------------- 1) cumsum of duration, mel_len ----------------
__global__ void cumsum_kernel(const float* __restrict__ dur,
                              float* __restrict__ sk, float* __restrict__ ek,
                              int* __restrict__ mlen, float* __restrict__ mel_out) {
    int b = threadIdx.x;
    if (b >= B_) return;
    float run = 0.f;
    for (int s = 0; s < TS_; ++s) {
        float d = dur[b * TS_ + s];
        run += d;
        ek[b * TS_ + s] = run;
        sk[b * TS_ + s] = run - d;
    }
    int ml = (int)rintf(run);
    if (ml > TM_) ml = TM_;
    if (ml < 0) ml = 0;
    mlen[b] = ml;
    mel_out[b] = (float)ml;
}

// ---------------- 2) ~mel_mask output ----------------
__global__ void mask_kernel(const int* __restrict__ mlen, float* __restrict__ mask_out) {
    int idx = blockIdx.x * blockDim.x + threadIdx.x;
    if (idx >= B_ * TM_) return;
    int b = idx / TM_, t = idx - b * TM_;
    mask_out[idx] = (t < mlen[b]) ? 1.f : 0.f;
}

// ---------------- 3) conv blocks -> Vw, Vc (wave per (b,s)) ----------------
__global__ void conv_kernel(const float* __restrict__ tok,
                            const float* __restrict__ wk, const float* __restrict__ wb,
                            const float* __restrict__ wbg, const float* __restrict__ wbb,
                            const float* __restrict__ wbm, const float* __restrict__ wbv,
                            const float* __restrict__ wlg, const float* __restrict__ wlb,
                            const float* __restrict__ ck, const float* __restrict__ cb,
                            const float* __restrict__ cbg, const float* __restrict__ cbb,
                            const float* __restrict__ cbm, const float* __restrict__ cbv,
                            const float* __restrict__ clg, const float* __restrict__ clb,
                            float* __restrict__ Vw, float* __restrict__ Vc) {
    int wid = (blockIdx.x * blockDim.x + threadIdx.x) >> 5;
    int lane = threadIdx.x & 31;
    if (wid >= B_ * TS_) return;
    int b = wid / TS_, s = wid - b * TS_;

    float aw[CO_], ac[CO_];
#pragma unroll
    for (int o = 0; o < CO_; ++o) { aw[o] = 0.f; ac[o] = 0.f; }

    for (int tap = 0; tap < 3; ++tap) {
        int ss = s + tap - 1;
        if (ss < 0 || ss >= TS_) continue;
        const float* tp = tok + ((size_t)b * TS_ + ss) * D_;
#pragma unroll
        for (int j = 0; j < D_ / 32; ++j) {
            int i = lane + 32 * j;
            float tv = tp[i];
            const float* kw = wk + ((size_t)tap * D_ + i) * CO_;
            const float* kc = ck + ((size_t)tap * D_ + i) * CO_;
#pragma unroll
            for (int o = 0; o < CO_; ++o) {
                aw[o] += tv * kw[o];
                ac[o] += tv * kc[o];
            }
        }
    }
#pragma unroll
    for (int m = 16; m >= 1; m >>= 1) {
#pragma unroll
        for (int o = 0; o < CO_; ++o) {
            aw[o] += __shfl_xor(aw[o], m, 32);
            ac[o] += __shfl_xor(ac[o], m, 32);
        }
    }
    if (lane == 0) {
        float yw[CO_], yc[CO_];
#pragma unroll
        for (int o = 0; o < CO_; ++o) {
            float y = aw[o] + wb[o];
            y = (y - wbm[o]) * rsqrtf(wbv[o] + 1e-5f) * wbg[o] + wbb[o];
            yw[o] = silu_f(y);
            float z = ac[o] + cb[o];
            z = (z - cbm[o]) * rsqrtf(cbv[o] + 1e-5f) * cbg[o] + cbb[o];
            yc[o] = silu_f(z);
        }
        float mw = 0.f, mc = 0.f;
#pragma unroll
        for (int o = 0; o < CO_; ++o) { mw += yw[o]; mc += yc[o]; }
        mw *= (1.f / CO_); mc *= (1.f / CO_);
        float vw = 0.f, vc = 0.f;
#pragma unroll
        for (int o = 0; o < CO_; ++o) {
            float ew = yw[o] - mw; vw += ew * ew;
            float ec = yc[o] - mc; vc += ec * ec;
        }
        float rw = rsqrtf(vw * (1.f / CO_) + 1e-5f);
        float rc = rsqrtf(vc * (1.f / CO_) + 1e-5f);
        float* ow = Vw + ((size_t)b * TS_ + s) * CO_;
        float* oc = Vc + ((size_t)b * TS_ + s) * CO_;
#pragma unroll
        for (int o = 0; o < CO_; ++o) {
            ow[o] = (yw[o] - mw) * rw * wlg[o] + wlb[o];
            oc[o] = (yc[o] - mc) * rc * clg[o] + clb[o];
        }
    }
}

// ---------------- 4) fused S/E, MLPs, softmax, Wp/C outputs, Bc ----------------
__global__ void __launch_bounds__(TS_)
swish_softmax_kernel(const float* __restrict__ sk, const float* __restrict__ ek,
                     const int* __restrict__ srclen, const int* __restrict__ mlen,
                     const float* __restrict__ Vw, const float* __restrict__ Vc,
                     const float* __restrict__ sw1, const float* __restrict__ sb1,
                     const float* __restrict__ sw2, const float* __restrict__ sb2,
                     const float* __restrict__ sw3, const float* __restrict__ sb3,
                     const float* __restrict__ c1, const float* __restrict__ cb1,
                     const float* __restrict__ c2, const float* __restrict__ cb2,
                     const float* __restrict__ c3, const float* __restrict__ cb3,
                     float* __restrict__ WpOut, float* __restrict__ COut,
                     float* __restrict__ BcOut) {
    __shared__ float red[8 * TS_];
    int idx = blockIdx.x;
    int b = idx / TM_, t = idx - b * TM_;
    int s = threadIdx.x;
    float tval = (float)(t + 1);
    bool smask = s >= srclen[b];
    int ml = mlen[b];
    bool tmask = t >= ml;
    bool amask = smask || tmask;

    float Sv = amask ? 0.f : (tval - sk[b * TS_ + s]);
    float Ev = amask ? 0.f : (ek[b * TS_ + s] - tval);

    float x[CO_ + 2];
    x[0] = Sv; x[1] = Ev;
    const float* vw = Vw + ((size_t)b * TS_ + s) * CO_;
#pragma unroll
    for (int i = 0; i < CO_; ++i) x[2 + i] = vw[i];

    // W-branch MLP: 10 -> 4 -> 4 -> 4  (uniform weight addresses -> scalar loads)
    float h1[DW_], h2[DW_], wr[DW_];
#pragma unroll
    for (int o = 0; o < DW_; ++o) {
        float a = sb1[o];
#pragma unroll
        for (int i = 0; i < CO_ + 2; ++i) a += x[i] * sw1[i * DW_ + o];
        h1[o] = silu_f(a);
    }
#pragma unroll
    for (int o = 0; o < DW_; ++o) {
        float a = sb2[o];
#pragma unroll
        for (int i = 0; i < DW_; ++i) a += h1[i] * sw2[i * DW_ + o];
        h2[o] = silu_f(a);
    }
    const float NEG_INF = -__builtin_huge_valf();
#pragma unroll
    for (int o = 0; o < DW_; ++o) {
        float a = sb3[o];
#pragma unroll
        for (int i = 0; i < DW_; ++i) a += h2[i] * sw3[i * DW_ + o];
        wr[o] = smask ? NEG_INF : a;
    }

    // masked softmax over s (axis=2), per q
#pragma unroll
    for (int q = 0; q < DW_; ++q) red[q * TS_ + s] = wr[q];
    __syncthreads();
    for (int off = TS_ / 2; off >= 1; off >>= 1) {
        if (s < off) {
#pragma unroll
            for (int q = 0; q < DW_; ++q)
                red[q * TS_ + s] = fmaxf(red[q * TS_ + s], red[q * TS_ + s + off]);
        }
        __syncthreads();
    }
    float mx[DW_];
#pragma unroll
    for (int q = 0; q < DW_; ++q) mx[q] = red[q * TS_];
    __syncthreads();

    float p[DW_];
#pragma unroll
    for (int q = 0; q < DW_; ++q) p[q] = smask ? 0.f : expf(wr[q] - mx[q]);
#pragma unroll
    for (int q = 0; q < DW_; ++q) red[q * TS_ + s] = p[q];
    __syncthreads();
    for (int off = TS_ / 2; off >= 1; off >>= 1) {
        if (s < off) {
#pragma unroll
            for (int q = 0; q < DW_; ++q)
                red[q * TS_ + s] += red[q * TS_ + s + off];
        }
        __syncthreads();
    }
    float sm[DW_];
#pragma unroll
    for (int q = 0; q < DW_; ++q) sm[q] = red[q * TS_];
    __syncthreads();

    float wsm[DW_];
#pragma unroll
    for (int q = 0; q < DW_; ++q) {
        float v = (sm[q] > 0.f) ? (p[q] / sm[q]) : 0.f;
        wsm[q] = tmask ? 0.f : v;
    }
#pragma unroll
    for (int q = 0; q < DW_; ++q)
        WpOut[(((size_t)b * DW_ + q) * TM_ + t) * TS_ + s] = wsm[q];

    // C-branch MLP: 10 -> 2 -> 2 -> 2
    const float* vc = Vc + ((size_t)b * TS_ + s) * CO_;
#pragma unroll
    for (int i = 0; i < CO_; ++i) x[2 + i] = vc[i];
    float g1[DC_], g2[DC_], cv[DC_];
#pragma unroll
    for (int o = 0; o < DC_; ++o) {
        float a = cb1[o];
#pragma unroll
        for (int i = 0; i < CO_ + 2; ++i) a += x[i] * c1[i * DC_ + o];
        g1[o] = silu_f(a);
    }
#pragma unroll
    for (int o = 0; o < DC_; ++o) {
        float a = cb2[o];
#pragma unroll
        for (int i = 0; i < DC_; ++i) a += g1[i] * c2[i * DC_ + o];
        g2[o] = silu_f(a);
    }
#pragma unroll
    for (int o = 0; o < DC_; ++o) {
        float a = cb3[o];
#pragma unroll
        for (int i = 0; i < DC_; ++i) a += g2[i] * c3[i * DC_ + o];
        cv[o] = a;
    }
    float* cout = COut + (((size_t)b * TM_ + t) * TS_ + s) * DC_;
    cout[0] = cv[0];
    cout[1] = cv[1];

    // Bc[t, q, p] = sum_s wsm[q] * cv[p]
#pragma unroll
    for (int k = 0; k < DW_ * DC_; ++k) red[k * TS_ + s] = wsm[k >> 1] * cv[k & 1];
    __syncthreads();
    for (int off = TS_ / 2; off >= 1; off >>= 1) {
        if (s < off) {
#pragma unroll
            for (int k = 0; k < DW_ * DC_; ++k)
                red[k * TS_ + s] += red[k * TS_ + s + off];
        }
        __syncthreads();
    }
    if (s < DW_ * DC_) BcOut[(size_t)idx * (DW_ * DC_) + s] = red[s * TS_];
}

// ---------------- WMMA f32 16x16x4 tile helper ----------------
__device__ __forceinline__ v8f wmma_tile_f32(const float* __restrict__ arow,
                                             const float* __restrict__ bcol, int ldb,
                                             int K, int lane) {
    const int half = lane >> 4;
    const int nl = lane & 15;
    v8f acc = {};
#pragma unroll 4
    for (int k0 = 0; k0 < K; k0 += 4) {
        v2f a, bm;
        a.x = arow[k0 + 2 * half];
        a.y = arow[k0 + 2 * half + 1];
        const float* bp = bcol + (size_t)(k0 + 2 * half) * ldb + nl;
        bm.x = bp[0];
        bm.y = bp[ldb];
        acc = __builtin_amdgcn_wmma_f32_16x16x4_f32(false, a, false, bm, (short)0, acc,
                                                    false, false);
    }
    return acc;
}

// ---------------- 5) GEMM1 per batch: A[(t, q*192+h)] = Wp[b,q,t,:] @ tokens[b] ----------------
__global__ void gemm1_kernel(const float* __restrict__ Wp, const float* __restrict__ tok,
                             float* __restrict__ Aout) {
    const int M = DW_ * TM_;         // 4000 (multiple of 16)
    const int TN = D_ / 16;          // 12
    int wid = (blockIdx.x * blockDim.x + threadIdx.x) >> 5;
    int lane = threadIdx.x & 31;
    int tm = wid / TN, tn = wid - tm * TN;
    if (tm >= M / 16) return;
    int m0 = tm * 16, n0 = tn * 16;
    int half = lane >> 4, nl = lane & 15;
    const float* arow = Wp + (size_t)(m0 + nl) * TS_;
    v8f acc = wmma_tile_f32(arow, tok + n0, D_, TS_, lane);
#pragma unroll
    for (int i = 0; i < 8; ++i) {
        int r = m0 + i + 8 * half;           // r = q*TM + t
        int q = r / TM_, t = r - q * TM_;
        Aout[(size_t)t * (DW_ * D_) + q * D_ + n0 + nl] = acc[i];
    }
}

// ---------------- 6) GEMM2 per batch: up = A@lw_W + lw_b + Bc@le_W ----------------
__global__ void gemm2_kernel(const float* __restrict__ A, const float* __restrict__ lwW,
                             const float* __restrict__ lwb, const float* __restrict__ Bc,
                             const float* __restrict__ leW, float* __restrict__ Up) {
    const int M = TM_;               // 1000
    const int K = DW_ * D_;          // 768
    const int TN = D_ / 16;          // 12
    int wid = (blockIdx.x * blockDim.x + threadIdx.x) >> 5;
    int lane = threadIdx.x & 31;
    int tm = wid / TN, tn = wid - tm * TN;
    if (tm >= (M + 15) / 16) return;
    int m0 = tm * 16, n0 = tn * 16;
    int half = lane >> 4, nl = lane & 15;
    int ar = m0 + nl; if (ar >= M) ar = M - 1;
    const float* arow = A + (size_t)ar * K;
    v8f acc = wmma_tile_f32(arow, lwW + n0, D_, K, lane);
#pragma unroll
    for (int i = 0; i < 8; ++i) {
        int r = m0 + i + 8 * half;
        if (r < M) {
            int c = n0 + nl;
            float v = acc[i] + lwb[c];
            const float* bc = Bc + (size_t)r * (DW_ * DC_);
#pragma unroll
            for (int j = 0; j < DW_ * DC_; ++j) v += bc[j] * leW[j * D_ + c];
            Up[(size_t)r * D_ + c] = v;
        }
    }
}

// ---------------- 7) LayerNorm over D + mel mask (wave per row) ----------------
__global__ void ln_kernel(float* __restrict__ up, const float* __restrict__ g,
                          const float* __restrict__ be, const int* __restrict__ mlen) {
    int wid = (blockIdx.x * blockDim.x + threadIdx.x) >> 5;
    int lane = threadIdx.x & 31;
    if (wid >= B_ * TM_) return;
    int b = wid / TM_, t = wid - b * TM_;
    float* row = up + (size_t)wid * D_;
    float x[D_ / 32];
    float s = 0.f;
#pragma unroll
    for (int j = 0; j < D_ / 32; ++j) { x[j] = row[lane + 32 * j]; s += x[j]; }
#pragma unroll
    for (int m = 16; m >= 1; m >>= 1) s += __shfl_xor(s, m, 32);
    float mean = s * (1.f / D_);
    float d = 0.f;
#pragma unroll
    for (int j = 0; j < D_ / 32; ++j) { float e = x[j] - mean; d += e * e; }
#pragma unroll
    for (int m = 16; m >= 1; m >>= 1) d += __shfl_xor(d, m, 32);
    float inv = rsqrtf(d * (1.f / D_) + 1e-5f);
    bool mm = t >= mlen[b];
#pragma unroll
    for (int j = 0; j < D_ / 32; ++j) {
        int c = lane + 32 * j;
        float y = (x[j] - mean) * inv * g[c] + be[c];
        row[c] = mm ? 0.f : y;
    }
}

// ---------------- 8) GEMM3: out = up @ po_W ----------------
__global__ void gemm3_kernel(const float* __restrict__ Up, const float* __restrict__ poW,
                             float* __restrict__ Out) {
    const int M = B_ * TM_;          // 16000 (multiple of 16)
    const int N = 2 * D_;            // 384
    const int TN = N / 16;           // 24
    int wid = (blockIdx.x * blockDim.x + threadIdx.x) >> 5;
    int lane = threadIdx.x & 31;
    int tm = wid / TN, tn = wid - tm * TN;
    if (tm >= M / 16) return;
    int m0 = tm * 16, n0 = tn * 16;
    int half = lane >> 4, nl = lane & 15;
    const float* arow = Up + (size_t)(m0 + nl) * D_;
    v8f acc = wmma_tile_f32(arow, poW + n0, N, D_, lane);
#pragma unroll
    for (int i = 0; i < 8; ++i)
        Out[(size_t)(m0 + i + 8 * half) * N + n0 + nl] = acc[i];
}

// ---------------- host ----------------
extern "C" void kernel_launch(void* const* d_in, const int* in_sizes, int n_in,
                              void* d_out, int out_size, void* d_ws, size_t ws_size,
                              hipStream_t stream) {
    const float* dur    = (const float*)d_in[0];
    const float* tok    = (const float*)d_in[1];
    const int*   srclen = (const int*)d_in[2];
    const float* cwk  = (const float*)d_in[6];
    const float* cwb  = (const float*)d_in[7];
    const float* cwbg = (const float*)d_in[8];
    const float* cwbb = (const float*)d_in[9];
    const float* cwbm = (const float*)d_in[10];
    const float* cwbv = (const float*)d_in[11];
    const float* cwlg = (const float*)d_in[12];
    const float* cwlb = (const float*)d_in[13];
    const float* sw1 = (const float*)d_in[14];
    const float* sb1 = (const float*)d_in[15];
    const float* sw2 = (const float*)d_in[16];
    const float* sb2 = (const float*)d_in[17];
    const float* sw3 = (const float*)d_in[18];
    const float* sb3 = (const float*)d_in[19];
    const float* lwW = (const float*)d_in[20];
    const float* lwb = (const float*)d_in[21];
    const float* cck  = (const float*)d_in[22];
    const float* ccb  = (const float*)d_in[23];
    const float* ccbg = (const float*)d_in[24];
    const float* ccbb = (const float*)d_in[25];
    const float* ccbm = (const float*)d_in[26];
    const float* ccbv = (const float*)d_in[27];
    const float* cclg = (const float*)d_in[28];
    const float* cclb = (const float*)d_in[29];
    const float* sc1 = (const float*)d_in[30];
    const float* scb1 = (const float*)d_in[31];
    const float* sc2 = (const float*)d_in[32];
    const float* scb2 = (const float*)d_in[33];
    const float* sc3 = (const float*)d_in[34];
    const float* scb3 = (const float*)d_in[35];
    const float* leW = (const float*)d_in[36];
    const float* lng = (const float*)d_in[37];
    const float* lnb = (const float*)d_in[38];
    const float* poW = (const float*)d_in[39];

    float* out = (float*)d_out;
    float* out_main = out + OUT_OFF;
    float* out_mask = out + MASK_OFF;
    float* out_mel  = out + MEL_OFF;
    float* out_wp   = out + WP_OFF;
    float* out_c    = out + C_OFF;

    // workspace carve-out (floats)
    float* ws = (float*)d_ws;
    float* sk   = ws;                         ws += B_ * TS_;
    float* ekv  = ws;                         ws += B_ * TS_;
    float* Vw   = ws;                         ws += B_ * TS_ * CO_;
    float* Vc   = ws;                         ws += B_ * TS_ * CO_;
    int*   mlen = (int*)ws;                   ws += 16;
    float* Bc   = ws;                         ws += B_ * TM_ * DW_ * DC_;
    float* Abuf = ws;                         ws += TM_ * DW_ * D_;      // per-batch
    float* Up   = ws;                         ws += B_ * TM_ * D_;

    // 1) cumsum + mel_len
    cumsum_kernel<<<1, 32, 0, stream>>>(dur, sk, ekv, mlen, out_mel);
    // 2) ~mel_mask
    mask_kernel<<<(B_ * TM_ + 255) / 256, 256, 0, stream>>>(mlen, out_mask);
    // 3) conv blocks (wave per (b,s): 8192 waves, 8 per block)
    conv_kernel<<<(B_ * TS_ + 7) / 8, 256, 0, stream>>>(
        tok, cwk, cwb, cwbg, cwbb, cwbm, cwbv, cwlg, cwlb,
        cck, ccb, ccbg, ccbb, ccbm, ccbv, cclg, cclb, Vw, Vc);
    // 4) fused swish/softmax -> Wp, C, Bc
    swish_softmax_kernel<<<B_ * TM_, TS_, 0, stream>>>(
        sk, ekv, srclen, mlen, Vw, Vc,
        sw1, sb1, sw2, sb2, sw3, sb3,
        sc1, scb1, sc2, scb2, sc3, scb3,
        out_wp, out_c, Bc);
    // 5/6) per-batch WMMA GEMMs (serialized on stream; Abuf recycled)
    for (int b = 0; b < B_; ++b) {
        const float* wp_b  = out_wp + (size_t)b * DW_ * TM_ * TS_;
        const float* tok_b = tok + (size_t)b * TS_ * D_;
        {
            int tiles = (DW_ * TM_ / 16) * (D_ / 16);     // 250*12 = 3000
            gemm1_kernel<<<(tiles + 3) / 4, 128, 0, stream>>>(wp_b, tok_b, Abuf);
        }
        {
            int tiles = ((TM_ + 15) / 16) * (D_ / 16);    // 63*12 = 756
            gemm2_kernel<<<(tiles + 3) / 4, 128, 0, stream>>>(
                Abuf, lwW, lwb, Bc + (size_t)b * TM_ * DW_ * DC_, leW,
                Up + (size_t)b * TM_ * D_);
        }
    }
    // 7) LayerNorm + mel mask (wave per row)
    ln_kernel<<<(B_ * TM_ + 7) / 8, 256, 0, stream>>>(Up, lng, lnb, mlen);
    // 8) out = up @ po_W
    {
        int tiles = (B_ * TM_ / 16) * (2 * D_ / 16);      // 1000*24 = 24000
        gemm3_kernel<<<(tiles + 3) / 4, 128, 0, stream>>>(Up, poW, out_main);
    }
}